// InvariantChamferLoss_16252156248267
// MI455X (gfx1250) — compile-verified
//
#include <hip/hip_runtime.h>

typedef __attribute__((ext_vector_type(2))) float v2f;
typedef __attribute__((ext_vector_type(8))) float v8f;
typedef __attribute__((ext_vector_type(4))) unsigned int v4u;
typedef __attribute__((ext_vector_type(4))) int v4i;
typedef __attribute__((ext_vector_type(8))) int v8i;

#define BLK 256
#define NN_COLS 4096   // target/source panel width staged in LDS (64 KB)

#if defined(__has_builtin)
#if __has_builtin(__builtin_amdgcn_tensor_load_to_lds) && \
    __has_builtin(__builtin_amdgcn_s_wait_tensorcnt)
#define HAVE_TDM 1
#endif
#endif
#ifndef HAVE_TDM
#define HAVE_TDM 0
#endif

// ---------------------------------------------------------------------------
// pack (x,y,z) -> (-2x,-2y,-2z,|p|^2)  for the WMMA B-matrix (bias folded in K=4)
// ---------------------------------------------------------------------------
__global__ void icp_pack_kernel(const float* __restrict__ pts,
                                float* __restrict__ packed, int total) {
  int i = blockIdx.x * blockDim.x + threadIdx.x;
  if (i >= total) return;
  float x = pts[i * 3 + 0], y = pts[i * 3 + 1], z = pts[i * 3 + 2];
  float4 o;
  o.x = -2.0f * x;
  o.y = -2.0f * y;
  o.z = -2.0f * z;
  o.w = x * x + y * y + z * z;
  ((float4*)packed)[i] = o;
}

__global__ void icp_copy_kernel(const float* __restrict__ src,
                                float* __restrict__ dst, int total) {
  int i = blockIdx.x * blockDim.x + threadIdx.x;
  if (i < total) dst[i] = src[i];
}

// ---------------------------------------------------------------------------
// per-batch source stats: mu_src (3), var_src (1).  Constant across iterations.
// ---------------------------------------------------------------------------
__global__ void icp_src_stats_kernel(const float* __restrict__ src,
                                     float* __restrict__ stats, int N) {
  int b = blockIdx.x, tid = threadIdx.x;
  const float* S = src + (size_t)b * N * 3;
  __shared__ float rx[BLK], ry[BLK], rz[BLK];
  __shared__ float mu[3];
  float sx = 0.f, sy = 0.f, sz = 0.f;
  for (int n = tid; n < N; n += BLK) {
    sx += S[n * 3 + 0]; sy += S[n * 3 + 1]; sz += S[n * 3 + 2];
  }
  rx[tid] = sx; ry[tid] = sy; rz[tid] = sz;
  __syncthreads();
  for (int s = BLK / 2; s > 0; s >>= 1) {
    if (tid < s) { rx[tid] += rx[tid + s]; ry[tid] += ry[tid + s]; rz[tid] += rz[tid + s]; }
    __syncthreads();
  }
  if (tid == 0) { mu[0] = rx[0] / N; mu[1] = ry[0] / N; mu[2] = rz[0] / N; }
  __syncthreads();
  float v = 0.f;
  for (int n = tid; n < N; n += BLK) {
    float dx = S[n * 3 + 0] - mu[0];
    float dy = S[n * 3 + 1] - mu[1];
    float dz = S[n * 3 + 2] - mu[2];
    v += dx * dx + dy * dy + dz * dz;
  }
  rx[tid] = v;
  __syncthreads();
  for (int s = BLK / 2; s > 0; s >>= 1) {
    if (tid < s) rx[tid] += rx[tid + s];
    __syncthreads();
  }
  if (tid == 0) {
    stats[b * 4 + 0] = mu[0]; stats[b * 4 + 1] = mu[1];
    stats[b * 4 + 2] = mu[2]; stats[b * 4 + 3] = rx[0];
  }
}

// ---------------------------------------------------------------------------
// WMMA nearest-neighbor / min-distance kernel, LDS-staged B panel.
//   A rows: points (x,y,z,1)             (16x4 fp32 fragment)
//   B cols: packed (-2x,-2y,-2z,|p|^2)   (4x16 fp32 fragment from LDS)
//   D = -2 a.b + |b|^2  -> argmin over cols == argmin of true sq-distance.
// The per-batch B panel (CB*16 bytes = 64 KB) is staged into LDS once per
// block via the Tensor Data Mover (fallback: cooperative b128 copy); the
// 256-tile sweep then reads conflict-free ds_load_b64 only.
// mode 0: write argmin index.  mode 1: write min distance (+|a|^2, clamped 0).
// Grid must give exactly B*(RA/16) waves; wave stays fully active for WMMA.
// ---------------------------------------------------------------------------
__global__ void __launch_bounds__(BLK)
icp_nn_wmma_kernel(const float* __restrict__ ptsA,   // (B,RA,3)
                   const float* __restrict__ packB,  // (B,CB,4), CB == NN_COLS
                   int RA,
                   int* __restrict__ outIdx,         // (B,RA)  mode 0
                   float* __restrict__ outMin,       // (B,RA)  mode 1
                   int mode) {
  const int lane = threadIdx.x & 31;
  const int wave = threadIdx.x >> 5;
  const int wavesPerBlock = blockDim.x >> 5;
  const int tilesPerBatch = RA >> 4;
  const int gtile = blockIdx.x * wavesPerBlock + wave;
  const int b = gtile / tilesPerBatch;
  const int r0 = (gtile % tilesPerBatch) << 4;

  const float* A  = ptsA  + (size_t)b * RA * 3;
  const float* Bp = packB + (size_t)b * NN_COLS * 4;

  __shared__ __align__(16) float tileLDS[NN_COLS * 4];   // 64 KB B panel

#if HAVE_TDM
  if (wave == 0) {
    // Tensor DMA descriptor (D#), groups 0/1 per CDNA5 ISA 8.3/8.4:
    // 1-D tile, data_size=8B, 8192 elements = whole 64 KB panel.
    unsigned long long ga = (unsigned long long)(uintptr_t)Bp;
    unsigned ldsOff = (unsigned)(uintptr_t)tileLDS;      // low 32 bits = LDS offset
    unsigned elems = (NN_COLS * 16u) / 8u;               // 8192 (8-byte units)
    v4u g0;
    g0[0] = 1u;                                          // count=1, user mode
    g0[1] = ldsOff;                                      // lds_addr
    g0[2] = (unsigned)ga;                                // global_addr[31:0]
    g0[3] = (unsigned)((ga >> 32) & 0x01FFFFFFu) | 0x80000000u; // addr[56:32] | type=2
    v8i g1;
    g1[0] = (int)(3u << 16);                             // data_size = 8 bytes
    g1[1] = (int)((elems & 0xFFFFu) << 16);              // tensor_dim0[15:0]
    g1[2] = (int)(((elems >> 16) & 0xFFFFu) | (1u << 16)); // tensor_dim0[31:16] | tensor_dim1=1
    g1[3] = (int)((elems & 0xFFFFu) << 16);              // tile_dim0 = elems (1-D tile)
    g1[4] = 0;                                           // tile_dim1=0, tile_dim2=0
    g1[5] = (int)elems;                                  // tensor_dim0_stride[31:0]
    g1[6] = 0;
    g1[7] = 0;
    v4i gz4 = {0, 0, 0, 0};
    v8i gz8 = {0, 0, 0, 0, 0, 0, 0, 0};
    // clang-23 / therock-10.0 signature: (v4u, v8i, v4i, v4i, v8i, i32 cpol)
    __builtin_amdgcn_tensor_load_to_lds(g0, g1, gz4, gz4, gz8, 0);
    __builtin_amdgcn_s_wait_tensorcnt(0);
  }
  __syncthreads();
#else
  {
    const float4* gsrc = (const float4*)Bp;
    float4* ldst = (float4*)tileLDS;
    for (int i = threadIdx.x; i < NN_COLS; i += BLK) ldst[i] = gsrc[i];
  }
  __syncthreads();
#endif

  const int col = lane & 15;   // row (A) / col (B,D) slot within tile
  const int hi  = lane >> 4;   // K-half selector

  // A fragment: lanes 0-15 -> (k0,k1) = (x,y); lanes 16-31 -> (k2,k3) = (z,1)
  const int arow = r0 + col;
  v2f afrag;
  {
    float ax = A[arow * 3 + 0], ay = A[arow * 3 + 1], az = A[arow * 3 + 2];
    if (hi == 0) { afrag.x = ax; afrag.y = ay; }
    else         { afrag.x = az; afrag.y = 1.0f; }
  }

  float best[8];
  int   bidx[8];
#pragma unroll
  for (int v = 0; v < 8; ++v) { best[v] = 3.4e38f; bidx[v] = 0; }

  const v2f* tileV = (const v2f*)tileLDS;
  const int laneSlot = (col << 1) | hi;   // v2f index of this lane's B half

#pragma unroll 2
  for (int c0 = 0; c0 < NN_COLS; c0 += 16) {
    v2f bfrag = tileV[(c0 << 1) + laneSlot];  // ds_load_b64, conflict-free
    v8f acc = {};
    acc = __builtin_amdgcn_wmma_f32_16x16x4_f32(
        false, afrag, false, bfrag, (short)0, acc, false, false);
    const int bc = c0 + col;
#pragma unroll
    for (int v = 0; v < 8; ++v) {
      float d = acc[v];
      if (d < best[v]) { best[v] = d; bidx[v] = bc; }
    }
  }

  // butterfly min/argmin across the 16 column-lanes of each half-wave
#pragma unroll
  for (int v = 0; v < 8; ++v) {
    float bv = best[v];
    int   bi = bidx[v];
    for (int s = 1; s < 16; s <<= 1) {
      float ov = __shfl_xor(bv, s, 32);
      int   oi = __shfl_xor(bi, s, 32);
      if (ov < bv || (ov == bv && oi < bi)) { bv = ov; bi = oi; }
    }
    best[v] = bv; bidx[v] = bi;
  }

  if (col == 0) {
    // lane 0 owns rows r0+0..7 (VGPR v -> row v); lane 16 owns rows r0+8..15
#pragma unroll
    for (int v = 0; v < 8; ++v) {
      int row = r0 + hi * 8 + v;
      if (mode == 0) {
        outIdx[(size_t)b * RA + row] = bidx[v];
      } else {
        float px = A[row * 3 + 0], py = A[row * 3 + 1], pz = A[row * 3 + 2];
        float d = best[v] + px * px + py * py + pz * pz;
        outMin[(size_t)b * RA + row] = d > 0.f ? d : 0.f;
      }
    }
  }
}

// ---------------------------------------------------------------------------
// per-batch correlation sums: sum(tgt_corr) (3) and sum(src (x) tgt_corr) (9)
// ---------------------------------------------------------------------------
__global__ void icp_corr_kernel(const float* __restrict__ src,
                                const float* __restrict__ tgt,
                                const int* __restrict__ idx,
                                float* __restrict__ sums, int N, int M) {
  int b = blockIdx.x, tid = threadIdx.x;
  const float* S = src + (size_t)b * N * 3;
  const float* T = tgt + (size_t)b * M * 3;
  const int*   I = idx + (size_t)b * N;
  float acc[12];
#pragma unroll
  for (int k = 0; k < 12; ++k) acc[k] = 0.f;
  for (int n = tid; n < N; n += BLK) {
    float sx = S[n * 3 + 0], sy = S[n * 3 + 1], sz = S[n * 3 + 2];
    int m = I[n];
    float tx = T[m * 3 + 0], ty = T[m * 3 + 1], tz = T[m * 3 + 2];
    acc[0] += tx; acc[1] += ty; acc[2] += tz;
    acc[3] += sx * tx; acc[4]  += sx * ty; acc[5]  += sx * tz;
    acc[6] += sy * tx; acc[7]  += sy * ty; acc[8]  += sy * tz;
    acc[9] += sz * tx; acc[10] += sz * ty; acc[11] += sz * tz;
  }
  __shared__ float red[BLK][12];
#pragma unroll
  for (int k = 0; k < 12; ++k) red[tid][k] = acc[k];
  __syncthreads();
  for (int s = BLK / 2; s > 0; s >>= 1) {
    if (tid < s) {
#pragma unroll
      for (int k = 0; k < 12; ++k) red[tid][k] += red[tid + s][k];
    }
    __syncthreads();
  }
  if (tid == 0) {
#pragma unroll
    for (int k = 0; k < 12; ++k) sums[b * 12 + k] = red[0][k];
  }
}

// ---------------------------------------------------------------------------
// Umeyama per batch: H from sums, 3x3 SVD via Jacobi of H^T H, then R, t, scale
// rts[b*13]: R row-major R[e][d] (9), t (3), scale (1)
// ---------------------------------------------------------------------------
__global__ void icp_umeyama_kernel(const float* __restrict__ sums,
                                   const float* __restrict__ stats,
                                   float* __restrict__ rts, int N, int Bn) {
  int b = threadIdx.x;
  if (b >= Bn) return;
  const float* sm = sums + b * 12;
  const float* st = stats + b * 4;
  float mus[3] = {st[0], st[1], st[2]};
  float var = st[3];
  float mut[3] = {sm[0] / N, sm[1] / N, sm[2] / N};
  float H[3][3];
  for (int d = 0; d < 3; ++d)
    for (int e = 0; e < 3; ++e)
      H[d][e] = sm[3 + d * 3 + e] - (float)N * mus[d] * mut[e];

  // A = H^T H (symmetric), Jacobi eigen -> V, sigma
  float A[3][3], V[3][3];
  for (int i = 0; i < 3; ++i)
    for (int j = 0; j < 3; ++j) {
      A[i][j] = H[0][i] * H[0][j] + H[1][i] * H[1][j] + H[2][i] * H[2][j];
      V[i][j] = (i == j) ? 1.f : 0.f;
    }
  const int PP[3] = {0, 0, 1}, QQ[3] = {1, 2, 2};
  for (int sweep = 0; sweep < 12; ++sweep) {
    for (int r = 0; r < 3; ++r) {
      int p = PP[r], q = QQ[r];
      float apq = A[p][q];
      if (fabsf(apq) < 1e-25f) continue;
      float tau = (A[q][q] - A[p][p]) / (2.f * apq);
      float tt = (tau >= 0.f) ? 1.f / (tau + sqrtf(1.f + tau * tau))
                              : 1.f / (tau - sqrtf(1.f + tau * tau));
      float c = 1.f / sqrtf(1.f + tt * tt);
      float s = tt * c;
      for (int k = 0; k < 3; ++k) {
        float akp = A[k][p], akq = A[k][q];
        A[k][p] = c * akp - s * akq;
        A[k][q] = s * akp + c * akq;
      }
      for (int k = 0; k < 3; ++k) {
        float apk = A[p][k], aqk = A[q][k];
        A[p][k] = c * apk - s * aqk;
        A[q][k] = s * apk + c * aqk;
      }
      for (int k = 0; k < 3; ++k) {
        float vkp = V[k][p], vkq = V[k][q];
        V[k][p] = c * vkp - s * vkq;
        V[k][q] = s * vkp + c * vkq;
      }
    }
  }
  float sig[3];
  for (int i = 0; i < 3; ++i) {
    float l = A[i][i];
    sig[i] = sqrtf(l > 0.f ? l : 0.f);
  }
  // sort descending (sigma + V columns)
  for (int i = 0; i < 2; ++i)
    for (int j = 0; j < 2 - i; ++j)
      if (sig[j] < sig[j + 1]) {
        float tsv = sig[j]; sig[j] = sig[j + 1]; sig[j + 1] = tsv;
        for (int k = 0; k < 3; ++k) {
          float tv = V[k][j]; V[k][j] = V[k][j + 1]; V[k][j + 1] = tv;
        }
      }
  // U columns: u_c = H v_c / |H v_c|;  last column falls back to cross product
  float U[3][3];
  for (int c = 0; c < 3; ++c) {
    float w0 = H[0][0] * V[0][c] + H[0][1] * V[1][c] + H[0][2] * V[2][c];
    float w1 = H[1][0] * V[0][c] + H[1][1] * V[1][c] + H[1][2] * V[2][c];
    float w2 = H[2][0] * V[0][c] + H[2][1] * V[1][c] + H[2][2] * V[2][c];
    float nw = sqrtf(w0 * w0 + w1 * w1 + w2 * w2);
    if (c == 2 && nw <= 1e-6f * (sig[0] + 1e-30f)) {
      w0 = U[1][0] * U[2][1] - U[2][0] * U[1][1];
      w1 = U[2][0] * U[0][1] - U[0][0] * U[2][1];
      w2 = U[0][0] * U[1][1] - U[1][0] * U[0][1];
      nw = sqrtf(w0 * w0 + w1 * w1 + w2 * w2);
    }
    float inv = (nw > 1e-30f) ? 1.f / nw : 0.f;
    U[0][c] = w0 * inv; U[1][c] = w1 * inv; U[2][c] = w2 * inv;
  }
  // d = sign(det(V) * det(U))   (columns)
  float detU = U[0][0] * (U[1][1] * U[2][2] - U[2][1] * U[1][2])
             - U[0][1] * (U[1][0] * U[2][2] - U[2][0] * U[1][2])
             + U[0][2] * (U[1][0] * U[2][1] - U[2][0] * U[1][1]);
  float detV = V[0][0] * (V[1][1] * V[2][2] - V[2][1] * V[1][2])
             - V[0][1] * (V[1][0] * V[2][2] - V[2][0] * V[1][2])
             + V[0][2] * (V[1][0] * V[2][1] - V[2][0] * V[1][1]);
  float dd = (detU * detV >= 0.f) ? 1.f : -1.f;

  float traceS = sig[0] + sig[1] + dd * sig[2];
  float scale = traceS / (var + 1e-8f);
  scale = fminf(fmaxf(scale, 0.5f), 2.0f);

  float R[3][3];  // R = V diag(1,1,d) U^T
  for (int i = 0; i < 3; ++i)
    for (int k = 0; k < 3; ++k)
      R[i][k] = V[i][0] * U[k][0] + V[i][1] * U[k][1] + dd * V[i][2] * U[k][2];

  float* o = rts + b * 13;
  for (int e = 0; e < 3; ++e)
    for (int d = 0; d < 3; ++d) o[e * 3 + d] = R[e][d];
  for (int e = 0; e < 3; ++e)
    o[9 + e] = mut[e] - scale * (mus[0] * R[e][0] + mus[1] * R[e][1] + mus[2] * R[e][2]);
  o[12] = scale;
}

// src_current = scale * source @ R^T + t
__global__ void icp_transform_kernel(const float* __restrict__ src,
                                     const float* __restrict__ rts,
                                     float* __restrict__ dst, int N, int total) {
  int i = blockIdx.x * blockDim.x + threadIdx.x;
  if (i >= total) return;
  int b = i / N;
  const float* p = src + (size_t)i * 3;
  const float* R = rts + b * 13;
  float s = R[12];
  float x = p[0], y = p[1], z = p[2];
  float* o = dst + (size_t)i * 3;
  o[0] = s * (x * R[0] + y * R[1] + z * R[2]) + R[9];
  o[1] = s * (x * R[3] + y * R[4] + z * R[5]) + R[10];
  o[2] = s * (x * R[6] + y * R[7] + z * R[8]) + R[11];
}

__global__ void icp_chamfer_batch_kernel(const float* __restrict__ mA,
                                         const float* __restrict__ mB,
                                         float* __restrict__ chamB, int N, int M) {
  int b = blockIdx.x, tid = threadIdx.x;
  float sa = 0.f, sb = 0.f;
  for (int n = tid; n < N; n += BLK) sa += mA[(size_t)b * N + n];
  for (int m = tid; m < M; m += BLK) sb += mB[(size_t)b * M + m];
  __shared__ float r1[BLK], r2[BLK];
  r1[tid] = sa; r2[tid] = sb;
  __syncthreads();
  for (int s = BLK / 2; s > 0; s >>= 1) {
    if (tid < s) { r1[tid] += r1[tid + s]; r2[tid] += r2[tid + s]; }
    __syncthreads();
  }
  if (tid == 0) chamB[b] = r1[0] / N + r2[0] / M;
}

__global__ void icp_finalize_kernel(const float* __restrict__ chamB,
                                    const float* __restrict__ aligned,
                                    float* __restrict__ out, int Bn, int total) {
  int i = blockIdx.x * blockDim.x + threadIdx.x;
  if (i < total) out[1 + i] = aligned[i];
  if (i == 0) {
    float s = 0.f;
    for (int b = 0; b < Bn; ++b) s += chamB[b];
    out[0] = s / (float)Bn;
  }
}

// ---------------------------------------------------------------------------
extern "C" void kernel_launch(void* const* d_in, const int* in_sizes, int n_in,
                              void* d_out, int out_size, void* d_ws, size_t ws_size,
                              hipStream_t stream) {
  (void)in_sizes; (void)n_in; (void)out_size; (void)ws_size;
  const int B = 16, N = 4096, M = 4096;
  const float* source = (const float*)d_in[0];
  const float* target = (const float*)d_in[1];
  float* out = (float*)d_out;

  float* w       = (float*)d_ws;
  float* packedT = w;                                  // B*M*4
  float* srcCur  = packedT + (size_t)B * M * 4;        // B*N*3
  float* packedS = srcCur  + (size_t)B * N * 3;        // B*N*4
  int*   idxBuf  = (int*)(packedS + (size_t)B * N * 4);// B*N
  float* sums    = (float*)(idxBuf + (size_t)B * N);   // B*12
  float* rts     = sums   + B * 12;                    // B*13
  float* stats   = rts    + B * 13;                    // B*4
  float* minS2T  = stats  + B * 4;                     // B*N
  float* minT2S  = minS2T + (size_t)B * N;             // B*M
  float* chamB   = minT2S + (size_t)B * M;             // B

  const int nnBlocks = B * (N / 16) / 8;  // 8 waves/block, 16 rows/wave

  icp_pack_kernel<<<(B * M + BLK - 1) / BLK, BLK, 0, stream>>>(target, packedT, B * M);
  icp_src_stats_kernel<<<B, BLK, 0, stream>>>(source, stats, N);
  icp_copy_kernel<<<(B * N * 3 + BLK - 1) / BLK, BLK, 0, stream>>>(source, srcCur, B * N * 3);

  for (int it = 0; it < 5; ++it) {
    icp_nn_wmma_kernel<<<nnBlocks, BLK, 0, stream>>>(srcCur, packedT, N,
                                                     idxBuf, nullptr, 0);
    icp_corr_kernel<<<B, BLK, 0, stream>>>(source, target, idxBuf, sums, N, M);
    icp_umeyama_kernel<<<1, 32, 0, stream>>>(sums, stats, rts, N, B);
    icp_transform_kernel<<<(B * N + BLK - 1) / BLK, BLK, 0, stream>>>(source, rts,
                                                                      srcCur, N, B * N);
  }

  // final chamfer: aligned source == srcCur after iteration 5
  icp_pack_kernel<<<(B * N + BLK - 1) / BLK, BLK, 0, stream>>>(srcCur, packedS, B * N);
  icp_nn_wmma_kernel<<<nnBlocks, BLK, 0, stream>>>(srcCur, packedT, N,
                                                   nullptr, minS2T, 1);
  icp_nn_wmma_kernel<<<B * (M / 16) / 8, BLK, 0, stream>>>(target, packedS, M,
                                                           nullptr, minT2S, 1);
  icp_chamfer_batch_kernel<<<B, BLK, 0, stream>>>(minS2T, minT2S, chamB, N, M);
  icp_finalize_kernel<<<(B * N * 3 + BLK - 1) / BLK, BLK, 0, stream>>>(
      chamB, srcCur, out, B, B * N * 3);
}